// GATEncoder_1494648619177
// MI455X (gfx1250) — compile-verified
//
#include <hip/hip_runtime.h>

typedef float v2f __attribute__((ext_vector_type(2)));
typedef float v8f __attribute__((ext_vector_type(8)));

constexpr int N    = 32768;
constexpr int E    = 524288;
constexpr int ET   = E + N;          // edges incl. self loops
constexpr int F_IN = 256;
constexpr int H    = 4;
constexpr int C    = 128;
constexpr int HC   = H * C;          // 512
constexpr float NEG_SLOPE = 0.2f;
constexpr float BN_EPS    = 1e-5f;

// ---------------------------------------------------------------------------
// WMMA f32 GEMM: Cout[M,Nc] = A[M,K] @ B[K,Nc] (+bias).
// Block = 8 waves = 128 rows x 64 cols.  The Kx64 B panel is staged once in
// LDS pre-swizzled into the exact V_WMMA_F32_16X16X4_F32 B-fragment layout:
//   Bs[(r/4)*256 + c*4 + (r&2) + (r&1)] = B[r][c]
// so each lane reads its (k,k+1) pair with one aligned ds_load_b64
// (lane c -> dwords {4c,4c+1}, lane c+16 -> {4c+2,4c+3}: all 64 banks,
// conflict-free).  Each wave: 16x64 tile, A fragment reused across 4 WMMAs.
// ---------------------------------------------------------------------------
__global__ void __launch_bounds__(256)
wmma_gemm_f32(const float* __restrict__ A,
              const float* __restrict__ B,
              float* __restrict__ Cout,
              const float* __restrict__ bias,
              int M, int K, int Nc) {
    extern __shared__ float Bs[];          // K * 64 floats, swizzled
    const int tilesN = Nc >> 6;
    const int bm = blockIdx.x / tilesN;    // 128-row block
    const int tn = blockIdx.x % tilesN;
    const int n0 = tn << 6;

    // stage B panel [K x 64] -> LDS in fragment order
    for (int idx = threadIdx.x; idx < K * 64; idx += 256) {
        const int r = idx >> 6, c = idx & 63;
        Bs[((r >> 2) << 8) + (c << 2) + (r & 2) + (r & 1)] = B[r * Nc + n0 + c];
    }
    __syncthreads();

    const int wave = threadIdx.x >> 5;
    const int lane = threadIdx.x & 31;
    const int m0 = (bm << 7) + (wave << 4);
    if (m0 >= M) return;

    v8f acc0 = {}, acc1 = {}, acc2 = {}, acc3 = {};

    // A 16x4 layout: lanes 0-15 rows M, VGPR0/1 = K0/K1; lanes 16-31 = K2/K3.
    const int hsel = lane >> 4;            // 0 (K0,K1) or 1 (K2,K3)
    const int col  = lane & 15;
    const float* __restrict__ Ap = A + (m0 + col) * K + (hsel << 1);
    const float* __restrict__ Bbase = Bs + (col << 2) + (hsel << 1);

    for (int k = 0; k < K; k += 4) {
        v2f a;
        a.x = Ap[k];
        a.y = Ap[k + 1];
        const float* __restrict__ Bp = Bbase + ((k >> 2) << 8);
        const v2f b0 = *(const v2f*)(Bp);        // col tile +0
        const v2f b1 = *(const v2f*)(Bp + 64);   // col tile +16
        const v2f b2 = *(const v2f*)(Bp + 128);  // col tile +32
        const v2f b3 = *(const v2f*)(Bp + 192);  // col tile +48
        acc0 = __builtin_amdgcn_wmma_f32_16x16x4_f32(false, a, false, b0, (short)0, acc0, false, false);
        acc1 = __builtin_amdgcn_wmma_f32_16x16x4_f32(false, a, false, b1, (short)0, acc1, false, false);
        acc2 = __builtin_amdgcn_wmma_f32_16x16x4_f32(false, a, false, b2, (short)0, acc2, false, false);
        acc3 = __builtin_amdgcn_wmma_f32_16x16x4_f32(false, a, false, b3, (short)0, acc3, false, false);
    }

    // C/D 16x16 layout: lanes 0-15 col N; VGPR r -> M=r (lo lanes) / M=8+r (hi)
    const int rowBase = m0 + (hsel << 3);
    float bv0 = 0.f, bv1 = 0.f, bv2 = 0.f, bv3 = 0.f;
    if (bias) {
        bv0 = bias[n0 + col];
        bv1 = bias[n0 + col + 16];
        bv2 = bias[n0 + col + 32];
        bv3 = bias[n0 + col + 48];
    }
#pragma unroll
    for (int r = 0; r < 8; ++r) {
        float* Crow = Cout + (rowBase + r) * Nc + n0 + col;
        Crow[0]  = acc0[r] + bv0;
        Crow[16] = acc1[r] + bv1;
        Crow[32] = acc2[r] + bv2;
        Crow[48] = acc3[r] + bv3;
    }
}

// ---------------------------------------------------------------------------
// Attention logits: a_src[n,h] = <h[n,h,:], att_src[h,:]>, same for dst.
// One wave per (n,h) row; wave32 shuffle reduction.
// ---------------------------------------------------------------------------
__global__ void att_scores(const float* __restrict__ hlin,
                           const float* __restrict__ att_s,
                           const float* __restrict__ att_d,
                           float* __restrict__ a_src,
                           float* __restrict__ a_dst) {
    const int wid  = blockIdx.x * (blockDim.x >> 5) + (threadIdx.x >> 5);
    const int lane = threadIdx.x & 31;
    if (wid >= N * H) return;
    const int n = wid / H, h = wid % H;
    const float* __restrict__ hp = hlin + n * HC + h * C;
    const float* __restrict__ sp = att_s + h * C;
    const float* __restrict__ dp = att_d + h * C;
    float ss = 0.f, sd = 0.f;
#pragma unroll
    for (int c = lane; c < C; c += 32) {
        const float v = hp[c];
        ss += v * sp[c];
        sd += v * dp[c];
    }
#pragma unroll
    for (int off = 16; off > 0; off >>= 1) {
        ss += __shfl_xor(ss, off, 32);
        sd += __shfl_xor(sd, off, 32);
    }
    if (lane == 0) { a_src[wid] = ss; a_dst[wid] = sd; }
}

// ---------------------------------------------------------------------------
// CSR-by-destination build (once per launch; shared by both GAT layers)
// ---------------------------------------------------------------------------
__device__ __forceinline__ int edge_dst(const long long* __restrict__ ei, int e) {
    return (e < E) ? (int)ei[E + e] : (e - E);
}
__device__ __forceinline__ int edge_src(const long long* __restrict__ ei, int e) {
    return (e < E) ? (int)ei[e] : (e - E);
}

__global__ void csr_count(const long long* __restrict__ ei, int* __restrict__ cnt) {
    const int e = blockIdx.x * blockDim.x + threadIdx.x;
    if (e >= ET) return;
    atomicAdd(&cnt[edge_dst(ei, e)], 1);
}

// single-block exclusive scan of cnt[N] -> rowptr[N+1]; 1024 thr x 32 elems
__global__ void __launch_bounds__(1024)
scan_rowptr(const int* __restrict__ cnt, int* __restrict__ rowptr) {
    __shared__ int tot[1024];
    const int t = threadIdx.x;
    const int base = t * 32;
    int s = 0;
#pragma unroll
    for (int i = 0; i < 32; ++i) s += cnt[base + i];
    tot[t] = s;
    __syncthreads();
    for (int off = 1; off < 1024; off <<= 1) {
        const int v = (t >= off) ? tot[t - off] : 0;
        __syncthreads();
        tot[t] += v;
        __syncthreads();
    }
    int run = (t == 0) ? 0 : tot[t - 1];   // exclusive prefix of this chunk
#pragma unroll
    for (int i = 0; i < 32; ++i) {
        rowptr[base + i] = run;
        run += cnt[base + i];
    }
    if (t == 1023) rowptr[N] = run;
}

__global__ void csr_fill(const long long* __restrict__ ei,
                         int* __restrict__ cursor, int* __restrict__ adj) {
    const int e = blockIdx.x * blockDim.x + threadIdx.x;
    if (e >= ET) return;
    const int pos = atomicAdd(&cursor[edge_dst(ei, e)], 1);
    adj[pos] = e;
}

// ---------------------------------------------------------------------------
// Fused GAT gather: one wave per destination node.  Segment softmax (max,
// denom) per head via lane-parallel edge loops + shuffle reductions, then a
// serial edge loop accumulating alpha*h[src] for all 4 heads x 128 channels
// in registers (4 acc/lane/head).  Writes head-mean + bias.  No float atomics;
// h stays L2-resident (64 MB << 192 MB L2).
// ---------------------------------------------------------------------------
__global__ void gat_gather(const long long* __restrict__ ei,
                           const int* __restrict__ rowptr,
                           const int* __restrict__ adj,
                           const float* __restrict__ a_src,
                           const float* __restrict__ a_dst,
                           const float* __restrict__ hlin,
                           const float* __restrict__ bias,
                           float* __restrict__ out) {
    const int d    = blockIdx.x * (blockDim.x >> 5) + (threadIdx.x >> 5);
    const int lane = threadIdx.x & 31;
    if (d >= N) return;
    const int rb = rowptr[d], re = rowptr[d + 1];

    const float ad0 = a_dst[d * H + 0], ad1 = a_dst[d * H + 1];
    const float ad2 = a_dst[d * H + 2], ad3 = a_dst[d * H + 3];

    // pass A: per-head segment max (lane-parallel over incident edges)
    float m0 = -1e30f, m1 = -1e30f, m2 = -1e30f, m3 = -1e30f;
    for (int j = rb + lane; j < re; j += 32) {
        const int s = edge_src(ei, adj[j]);
        const float* as = a_src + s * H;
        float v;
        v = as[0] + ad0; v = v > 0.f ? v : NEG_SLOPE * v; m0 = fmaxf(m0, v);
        v = as[1] + ad1; v = v > 0.f ? v : NEG_SLOPE * v; m1 = fmaxf(m1, v);
        v = as[2] + ad2; v = v > 0.f ? v : NEG_SLOPE * v; m2 = fmaxf(m2, v);
        v = as[3] + ad3; v = v > 0.f ? v : NEG_SLOPE * v; m3 = fmaxf(m3, v);
    }
#pragma unroll
    for (int off = 16; off > 0; off >>= 1) {
        m0 = fmaxf(m0, __shfl_xor(m0, off, 32));
        m1 = fmaxf(m1, __shfl_xor(m1, off, 32));
        m2 = fmaxf(m2, __shfl_xor(m2, off, 32));
        m3 = fmaxf(m3, __shfl_xor(m3, off, 32));
    }

    // pass B: per-head softmax denominator
    float s0 = 0.f, s1 = 0.f, s2 = 0.f, s3 = 0.f;
    for (int j = rb + lane; j < re; j += 32) {
        const int s = edge_src(ei, adj[j]);
        const float* as = a_src + s * H;
        float v;
        v = as[0] + ad0; v = v > 0.f ? v : NEG_SLOPE * v; s0 += __expf(v - m0);
        v = as[1] + ad1; v = v > 0.f ? v : NEG_SLOPE * v; s1 += __expf(v - m1);
        v = as[2] + ad2; v = v > 0.f ? v : NEG_SLOPE * v; s2 += __expf(v - m2);
        v = as[3] + ad3; v = v > 0.f ? v : NEG_SLOPE * v; s3 += __expf(v - m3);
    }
#pragma unroll
    for (int off = 16; off > 0; off >>= 1) {
        s0 += __shfl_xor(s0, off, 32);
        s1 += __shfl_xor(s1, off, 32);
        s2 += __shfl_xor(s2, off, 32);
        s3 += __shfl_xor(s3, off, 32);
    }
    const float r0 = 1.f / (s0 + 1e-16f), r1 = 1.f / (s1 + 1e-16f);
    const float r2 = 1.f / (s2 + 1e-16f), r3 = 1.f / (s3 + 1e-16f);

    // pass C: serial over edges, lanes parallel over channels.
    // acc[h][q]: channel c = q*32 + lane of head h.
    float acc[H][4];
#pragma unroll
    for (int h = 0; h < H; ++h)
#pragma unroll
        for (int q = 0; q < 4; ++q) acc[h][q] = 0.f;

    for (int j = rb; j < re; ++j) {
        const int s = edge_src(ei, adj[j]);
        const float* as = a_src + s * H;    // wave-uniform loads
        float v, w0, w1, w2, w3;
        v = as[0] + ad0; v = v > 0.f ? v : NEG_SLOPE * v; w0 = __expf(v - m0) * r0;
        v = as[1] + ad1; v = v > 0.f ? v : NEG_SLOPE * v; w1 = __expf(v - m1) * r1;
        v = as[2] + ad2; v = v > 0.f ? v : NEG_SLOPE * v; w2 = __expf(v - m2) * r2;
        v = as[3] + ad3; v = v > 0.f ? v : NEG_SLOPE * v; w3 = __expf(v - m3) * r3;
        const float* __restrict__ hp = hlin + s * HC + lane;
#pragma unroll
        for (int q = 0; q < 4; ++q) {
            acc[0][q] += w0 * hp[0 * C + q * 32];
            acc[1][q] += w1 * hp[1 * C + q * 32];
            acc[2][q] += w2 * hp[2 * C + q * 32];
            acc[3][q] += w3 * hp[3 * C + q * 32];
        }
    }

    // head mean + bias
#pragma unroll
    for (int q = 0; q < 4; ++q) {
        const int c = q * 32 + lane;
        out[d * C + c] = 0.25f * (acc[0][q] + acc[1][q] + acc[2][q] + acc[3][q]) + bias[c];
    }
}

// per-channel batch-norm statistics (block per channel)
__global__ void bn_stats(const float* __restrict__ x,
                         float* __restrict__ mu,
                         float* __restrict__ var) {
    const int c = blockIdx.x;
    float s = 0.f, s2 = 0.f;
    for (int n = threadIdx.x; n < N; n += blockDim.x) {
        const float v = x[n * C + c];
        s += v; s2 += v * v;
    }
    __shared__ float sb[256], s2b[256];
    sb[threadIdx.x] = s; s2b[threadIdx.x] = s2;
    __syncthreads();
    for (int off = blockDim.x >> 1; off > 0; off >>= 1) {
        if ((int)threadIdx.x < off) {
            sb[threadIdx.x]  += sb[threadIdx.x + off];
            s2b[threadIdx.x] += s2b[threadIdx.x + off];
        }
        __syncthreads();
    }
    if (threadIdx.x == 0) {
        const float m = sb[0] * (1.0f / N);
        mu[c]  = m;
        var[c] = s2b[0] * (1.0f / N) - m * m;
    }
}

__global__ void bn_relu(const float* __restrict__ x,
                        const float* __restrict__ mu,
                        const float* __restrict__ var,
                        const float* __restrict__ g,
                        const float* __restrict__ be,
                        float* __restrict__ y) {
    const int t = blockIdx.x * blockDim.x + threadIdx.x;
    if (t >= N * C) return;
    const int c = t % C;
    const float v = (x[t] - mu[c]) * rsqrtf(var[c] + BN_EPS) * g[c] + be[c];
    y[t] = v > 0.f ? v : 0.f;
}

// ---------------------------------------------------------------------------
static inline int cdiv(int a, int b) { return (a + b - 1) / b; }

extern "C" void kernel_launch(void* const* d_in, const int* in_sizes, int n_in,
                              void* d_out, int out_size, void* d_ws, size_t ws_size,
                              hipStream_t stream) {
    (void)in_sizes; (void)n_in; (void)out_size; (void)ws_size;

    const float*     x  = (const float*)d_in[0];
    const long long* ei = (const long long*)d_in[1];
    const float* W1  = (const float*)d_in[2];
    const float* as1 = (const float*)d_in[3];
    const float* ad1 = (const float*)d_in[4];
    const float* b1  = (const float*)d_in[5];
    const float* g1  = (const float*)d_in[6];
    const float* be1 = (const float*)d_in[7];
    const float* W2  = (const float*)d_in[8];
    const float* as2 = (const float*)d_in[9];
    const float* ad2 = (const float*)d_in[10];
    const float* b2  = (const float*)d_in[11];
    const float* g2  = (const float*)d_in[12];
    const float* be2 = (const float*)d_in[13];
    const float* Wf  = (const float*)d_in[14];
    const float* bf  = (const float*)d_in[15];
    float* out = (float*)d_out;

    // workspace layout
    char* base = (char*)d_ws;
    size_t off = 0;
    auto take = [&](size_t bytes) {
        char* p = base + off;
        off += (bytes + 255) & ~(size_t)255;
        return p;
    };
    float* h_lin   = (float*)take((size_t)N * HC * 4);     // 64 MB
    float* nodebuf = (float*)take((size_t)N * C * 4);      // 16 MB
    float* xnode   = (float*)take((size_t)N * C * 4);      // 16 MB
    float* a_src   = (float*)take((size_t)N * H * 4);
    float* a_dst   = (float*)take((size_t)N * H * 4);
    int*   cnt     = (int*)  take((size_t)N * 4);
    int*   rowptr  = (int*)  take((size_t)(N + 1) * 4);
    int*   cursor  = (int*)  take((size_t)N * 4);
    int*   adj     = (int*)  take((size_t)ET * 4);         // ~2.2 MB
    float* mu      = (float*)take((size_t)C * 4);
    float* var     = (float*)take((size_t)C * 4);

    const int TB = 256;
    const int edgeBlocks = cdiv(ET, TB);
    const int nodeBlocks = cdiv(N * C, TB);
    const int attBlocks  = cdiv(N * H, TB / 32);
    const int dstBlocks  = cdiv(N, TB / 32);

    // ---- build CSR by destination (shared by both layers) ----
    hipMemsetAsync(cnt, 0, (size_t)N * 4, stream);
    csr_count<<<edgeBlocks, TB, 0, stream>>>(ei, cnt);
    scan_rowptr<<<1, 1024, 0, stream>>>(cnt, rowptr);
    hipMemcpyAsync(cursor, rowptr, (size_t)N * 4, hipMemcpyDeviceToDevice, stream);
    csr_fill<<<edgeBlocks, TB, 0, stream>>>(ei, cursor, adj);

    auto gat_layer = [&](const float* xin, int Din, const float* W,
                         const float* as_, const float* ad_, const float* b,
                         const float* g, const float* be, float* xout) {
        wmma_gemm_f32<<<(N / 128) * (HC / 64), TB, (size_t)Din * 64 * 4, stream>>>(
            xin, W, h_lin, nullptr, N, Din, HC);
        att_scores<<<attBlocks, TB, 0, stream>>>(h_lin, as_, ad_, a_src, a_dst);
        gat_gather<<<dstBlocks, TB, 0, stream>>>(ei, rowptr, adj, a_src, a_dst,
                                                 h_lin, b, nodebuf);
        bn_stats<<<C, TB, 0, stream>>>(nodebuf, mu, var);
        bn_relu<<<nodeBlocks, TB, 0, stream>>>(nodebuf, mu, var, g, be, xout);
    };

    gat_layer(x,     F_IN, W1, as1, ad1, b1, g1, be1, xnode);
    gat_layer(xnode, C,    W2, as2, ad2, b2, g2, be2, xnode);

    // final linear: out = xnode @ Wf + bf
    wmma_gemm_f32<<<(N / 128) * (C / 64), TB, (size_t)C * 64 * 4, stream>>>(
        xnode, Wf, out, bf, N, C, C);
}